// graphattention_layer_40218073759786
// MI455X (gfx1250) — compile-verified
//
#include <hip/hip_runtime.h>
#include <math.h>

#define N_NODES 10000
#define D_IN    256
#define D_OUT   128

typedef float v2f __attribute__((ext_vector_type(2)));
typedef float v8f __attribute__((ext_vector_type(8)));

// -------- Phase 1: WH = relu(H @ W^T + b) via v_wmma_f32_16x16x4_f32 --------
// Grid: 625 blocks (16 rows each) x 256 threads (8 waves, one 16-col tile per wave).
__global__ __launch_bounds__(256) void wh_wmma_kernel(const float* __restrict__ H,
                                                      const float* __restrict__ W,
                                                      const float* __restrict__ bias,
                                                      float* __restrict__ WH) {
  const int wave = threadIdx.x >> 5;
  const int lane = threadIdx.x & 31;
  const int half = lane >> 4;     // 0: lanes 0-15, 1: lanes 16-31
  const int lr   = lane & 15;
  const int row0 = blockIdx.x * 16;
  const int n0   = wave * 16;

  // A-frag (16x4 f32): lane L holds row M=L%16; v0 = K+2*half, v1 = K+2*half+1
  const float* hrow = H + (size_t)(row0 + lr) * D_IN + 2 * half;
  // B-frag (4x16 f32): B[k][n] = W^T[k][n] = W[n][k]; same striping, n = lr
  const float* wrow = W + (size_t)(n0 + lr) * D_IN + 2 * half;

  v8f c = {0.f, 0.f, 0.f, 0.f, 0.f, 0.f, 0.f, 0.f};
  for (int k = 0; k < D_IN; k += 4) {
    v2f a, b;
    a.x = hrow[k];
    a.y = hrow[k + 1];
    b.x = wrow[k];
    b.y = wrow[k + 1];
    c = __builtin_amdgcn_wmma_f32_16x16x4_f32(false, a, false, b, (short)0, c, false, false);
  }

  const float bv = bias[n0 + lr];
  for (int v = 0; v < 8; ++v) {
    const int row = row0 + v + 8 * half;     // C/D layout: VGPR v -> M=v (lo half), v+8 (hi half)
    float val = c[v] + bv;
    val = val > 0.f ? val : 0.f;
    WH[(size_t)row * D_OUT + n0 + lr] = val;
  }
}

// -------- Phase 2a: sender/receiver scores --------
__global__ __launch_bounds__(256) void scores_kernel(const float* __restrict__ WH,
                                                     const float* __restrict__ vt_w,
                                                     const float* __restrict__ vt_b,
                                                     const float* __restrict__ vr_w,
                                                     const float* __restrict__ vr_b,
                                                     float* __restrict__ s,
                                                     float* __restrict__ r) {
  const int row = blockIdx.x * blockDim.x + threadIdx.x;
  if (row >= N_NODES) return;
  const float* wh = WH + (size_t)row * D_OUT;
  float ds = 0.f, dr = 0.f;
  for (int k = 0; k < D_OUT; ++k) {
    const float v = wh[k];
    ds += v * vt_w[k];
    dr += v * vr_w[k];
  }
  s[row] = ds + vt_b[0];
  r[row] = dr + vr_b[0];
}

// -------- Phase 2b: deterministic column sum of WH (S_all = sum_j WH[j]) --------
#define NPART 64
__global__ __launch_bounds__(128) void colsum_partial_kernel(const float* __restrict__ WH,
                                                             float* __restrict__ partial) {
  const int p = blockIdx.x;   // 0..NPART-1
  const int n = threadIdx.x;  // 0..127
  const int rows_per = (N_NODES + NPART - 1) / NPART;
  const int r0 = p * rows_per;
  const int r1 = (r0 + rows_per < N_NODES) ? (r0 + rows_per) : N_NODES;
  float acc = 0.f;
  for (int row = r0; row < r1; ++row) acc += WH[(size_t)row * D_OUT + n];
  partial[p * D_OUT + n] = acc;
}

__global__ __launch_bounds__(128) void colsum_final_kernel(const float* __restrict__ partial,
                                                           float* __restrict__ S_all) {
  const int n = threadIdx.x;
  float acc = 0.f;
  for (int p = 0; p < NPART; ++p) acc += partial[p * D_OUT + n];
  S_all[n] = acc;
}

// -------- Phase 3: sparse-corrected softmax-attention apply --------
#define TILE_J 256
#define ROWS_PER_BLOCK 32
#define AT_STRIDE 36              // 144B rows: 16B-aligned for async b128; 2-way bank conflict max
#define E05 1.64872127070012814685f   // exp(0.5)

__device__ __forceinline__ unsigned wave_ballot32(bool p) {
#if __has_builtin(__builtin_amdgcn_ballot_w32)
  return __builtin_amdgcn_ballot_w32(p);
#else
  return (unsigned)__ballot(p);
#endif
}

// Async copy 16B global -> LDS via the gfx1250 async-LDS engine (ASYNCcnt-tracked).
// VDST operand is a per-lane LDS byte offset; on AMDGPU, flat pointers to LDS carry
// the LDS offset in their low 32 bits, so truncation yields the correct offset.
__device__ __forceinline__ void async_load_b128(void* lds_ptr, const void* gptr) {
  unsigned loff = (unsigned)(uintptr_t)lds_ptr;
  asm volatile("global_load_async_to_lds_b128 %0, %1, off"
               :: "v"(loff), "v"(gptr)
               : "memory");
}

__device__ __forceinline__ void wait_asynccnt0() {
#if __has_builtin(__builtin_amdgcn_s_wait_asynccnt)
  __builtin_amdgcn_s_wait_asynccnt(0);
#else
  asm volatile("s_wait_asynccnt 0x0" ::: "memory");
#endif
}

// Stage one 256x32 transposed tile of A (At[jr][ic] = A[J0+jr, I0+ic]) plus s[J0..].
__device__ __forceinline__ void stage_tile(const float* __restrict__ A,
                                           const float* __restrict__ s,
                                           int J0, int I0, int tid,
                                           float (*At)[AT_STRIDE], float* s_tile,
                                           bool use_async) {
  s_tile[tid] = (J0 + tid < N_NODES) ? s[J0 + tid] : 0.f;
  if (use_async) {
    // 2048 16B-quads / 256 threads = 8 async b128 per thread; 8 consecutive lanes
    // cover one 128B row segment of A -> coalesced global, aligned LDS dest.
    for (int p = 0; p < 8; ++p) {
      const int q   = p * 256 + tid;      // 0..2047
      const int jr  = q >> 3;
      const int ic4 = (q & 7) * 4;
      async_load_b128(&At[jr][ic4], A + (size_t)(J0 + jr) * N_NODES + I0 + ic4);
    }
  } else {
    // guarded scalar path for boundary tiles
    for (int p = 0; p < (TILE_J * ROWS_PER_BLOCK) / 256; ++p) {
      const int idx = tid + p * 256;
      const int jr  = idx >> 5;
      const int ic  = idx & 31;
      const int j   = J0 + jr;
      const int i   = I0 + ic;
      At[jr][ic] = (j < N_NODES && i < N_NODES) ? A[(size_t)j * N_NODES + i] : 0.f;
    }
  }
}

__global__ __launch_bounds__(256) void attention_kernel(const float* __restrict__ A,
                                                        const float* __restrict__ WH,
                                                        const float* __restrict__ s,
                                                        const float* __restrict__ r,
                                                        const float* __restrict__ S_all,
                                                        float* __restrict__ out) {
  __shared__ float At[2][TILE_J][AT_STRIDE];  // double-buffered transposed A tiles
  __shared__ float s_tile[2][TILE_J];

  const int tid  = threadIdx.x;
  const int wave = tid >> 5;
  const int lane = tid & 31;
  const int I0   = blockIdx.x * ROWS_PER_BLOCK;
  const bool interiorI = (I0 + ROWS_PER_BLOCK <= N_NODES);

  // each of the 8 waves owns 4 consecutive rows
  int   irow[4];
  float rloc[4];
  float Zacc[4] = {0.f, 0.f, 0.f, 0.f};
  float acc[4][4] = {};              // lane owns columns lane*4 .. lane*4+3
  for (int rr = 0; rr < 4; ++rr) {
    irow[rr] = I0 + wave * 4 + rr;
    rloc[rr] = (irow[rr] < N_NODES) ? r[irow[rr]] : 0.f;
  }

  const int nTiles = (N_NODES + TILE_J - 1) / TILE_J;

  // prologue: stage tile 0 into buffer 0
  stage_tile(A, s, 0, I0, tid, At[0], s_tile[0], interiorI && (TILE_J <= N_NODES));
  wait_asynccnt0();
  __syncthreads();

  for (int t = 0; t < nTiles; ++t) {
    const int b  = t & 1;
    const int J0 = t * TILE_J;

    // prefetch next tile into the other buffer (async engine overlaps with compute)
    if (t + 1 < nTiles) {
      const int J1 = (t + 1) * TILE_J;
      stage_tile(A, s, J1, I0, tid, At[b ^ 1], s_tile[b ^ 1],
                 interiorI && (J1 + TILE_J <= N_NODES));
    }

    for (int rr = 0; rr < 4; ++rr) {
      const int  i  = irow[rr];
      const bool vi = (i < N_NODES);
      const float* arow = A + (size_t)i * N_NODES;
      const float rv = rloc[rr];
      for (int tt = 0; tt < TILE_J / 32; ++tt) {
        const int jc = tt * 32 + lane;
        const int j  = J0 + jc;
        const float a1 = (vi && j < N_NODES) ? arow[j] : 0.f;   // A[i,j], coalesced 128B
        const float a2 = At[b][jc][i - I0];                     // A[j,i] from LDS
        const float m  = a1 * s_tile[b][jc] + a2 * rv;
        float delta = 0.f;
        if (m != 0.f) {
          const float sig = 1.f / (1.f + __expf(-m));
          delta = __expf(sig) - E05;
          Zacc[rr] += delta;
        }
        unsigned msk = wave_ballot32(delta != 0.f);
        while (msk) {
          const int l = __builtin_ctz(msk);
          msk &= msk - 1;
          const float ds = __shfl(delta, l, 32);
          const int   js = J0 + tt * 32 + l;
          const float4 w4 = ((const float4*)WH)[(size_t)js * (D_OUT / 4) + lane];
          acc[rr][0] += ds * w4.x;
          acc[rr][1] += ds * w4.y;
          acc[rr][2] += ds * w4.z;
          acc[rr][3] += ds * w4.w;
        }
      }
    }

    wait_asynccnt0();   // next tile's async LDS writes complete (per-wave)
    __syncthreads();    // all waves done writing next buffer / reading current buffer
  }

  const float4 sall = ((const float4*)S_all)[lane];
  for (int rr = 0; rr < 4; ++rr) {
    float z = Zacc[rr];
    for (int off = 16; off; off >>= 1) z += __shfl_xor(z, off, 32);
    const float Z   = (float)N_NODES * E05 + z;
    const float inv = 1.f / Z;
    const int i = irow[rr];
    if (i < N_NODES) {
      float4 o;
      o.x = (E05 * sall.x + acc[rr][0]) * inv;
      o.y = (E05 * sall.y + acc[rr][1]) * inv;
      o.z = (E05 * sall.z + acc[rr][2]) * inv;
      o.w = (E05 * sall.w + acc[rr][3]) * inv;
      ((float4*)out)[(size_t)i * (D_OUT / 4) + lane] = o;
    }
  }
}

// ---------------------------------------------------------------------------
extern "C" void kernel_launch(void* const* d_in, const int* in_sizes, int n_in,
                              void* d_out, int out_size, void* d_ws, size_t ws_size,
                              hipStream_t stream) {
  const float* H_k   = (const float*)d_in[0];
  const float* A     = (const float*)d_in[1];
  const float* W_w   = (const float*)d_in[2];
  const float* W_b   = (const float*)d_in[3];
  const float* vkt_w = (const float*)d_in[4];
  const float* vkt_b = (const float*)d_in[5];
  const float* vkr_w = (const float*)d_in[6];
  const float* vkr_b = (const float*)d_in[7];
  float* out = (float*)d_out;

  // workspace carve-up (256B aligned)
  char* ws = (char*)d_ws;
  float* WH      = (float*)(ws + 0);                 // 10000*128*4 = 5,120,000
  float* s_buf   = (float*)(ws + 5120000);           // 40,000 (pad to 40,960)
  float* r_buf   = (float*)(ws + 5160960);           // 40,960
  float* partial = (float*)(ws + 5201920);           // 64*128*4 = 32,768
  float* S_all   = (float*)(ws + 5234688);           // 512
  (void)ws_size; (void)in_sizes; (void)n_in; (void)out_size;

  // Phase 1: WH = relu(H @ W^T + b)  (WMMA f32)
  wh_wmma_kernel<<<N_NODES / 16, 256, 0, stream>>>(H_k, W_w, W_b, WH);

  // Phase 2: scores + column sum
  scores_kernel<<<(N_NODES + 255) / 256, 256, 0, stream>>>(WH, vkt_w, vkt_b, vkr_w, vkr_b,
                                                           s_buf, r_buf);
  colsum_partial_kernel<<<NPART, 128, 0, stream>>>(WH, partial);
  colsum_final_kernel<<<1, 128, 0, stream>>>(partial, S_all);

  // Phase 3: fused sparse-corrected softmax attention apply (async double-buffered tiles)
  attention_kernel<<<(N_NODES + ROWS_PER_BLOCK - 1) / ROWS_PER_BLOCK, 256, 0, stream>>>(
      A, WH, s_buf, r_buf, S_all, out);
}